// CRANUnit_37641093382854
// MI455X (gfx1250) — compile-verified
//
#include <hip/hip_runtime.h>
#include <hip/hip_bf16.h>
#include <math.h>

// Problem constants from the reference
#define Bb 32
#define Ee 1024
#define Hh 1024
#define Nn 128
#define Ll 64
#define Kk 8

typedef __attribute__((ext_vector_type(2))) float v2f;
typedef __attribute__((ext_vector_type(8))) float v8f;

__device__ __forceinline__ float wave_reduce_sum(float v) {
#pragma unroll
    for (int off = 16; off > 0; off >>= 1) v += __shfl_xor(v, off, 32);
    return v;
}

// -------------------------------------------------------------------------
// Kernel 1: scores[b,n] = cache_keys[b,n,:] . inputs[b,:]; top-8 + softmax
// -------------------------------------------------------------------------
__global__ void k_scores_topk(const float* __restrict__ inp,
                              const float* __restrict__ ck,
                              float* __restrict__ wts,
                              int* __restrict__ tidx) {
    __shared__ float s_in[Ee];
    __shared__ float s_sc[Nn];
    const int b = blockIdx.x;
    for (int e = threadIdx.x; e < Ee; e += 256) s_in[e] = inp[b * Ee + e];
    __syncthreads();

    const int lane = threadIdx.x & 31;
    const int wave = threadIdx.x >> 5;
    for (int n = wave; n < Nn; n += 8) {
        const float* row = ck + ((size_t)b * Nn + n) * Ee;
        float p = 0.f;
#pragma unroll
        for (int j = 0; j < 8; ++j) {
            const int e = lane * 4 + j * 128;
            const float4 v = *(const float4*)(row + e);
            p += v.x * s_in[e] + v.y * s_in[e + 1] + v.z * s_in[e + 2] + v.w * s_in[e + 3];
        }
        p = wave_reduce_sum(p);
        if (lane == 0) s_sc[n] = p;
    }
    __syncthreads();

    if (threadIdx.x == 0) {
        float tv[Kk]; int ti[Kk];
        for (int k = 0; k < Kk; ++k) {
            float best = -INFINITY; int bi = 0;
            for (int n = 0; n < Nn; ++n)
                if (s_sc[n] > best) { best = s_sc[n]; bi = n; }
            tv[k] = best; ti[k] = bi; s_sc[bi] = -INFINITY;
        }
        const float m = tv[0];
        float s = 0.f, ev[Kk];
        for (int k = 0; k < Kk; ++k) { ev[k] = expf(tv[k] - m); s += ev[k]; }
        for (int k = 0; k < Kk; ++k) {
            wts[b * Kk + k] = ev[k] / s;
            tidx[b * Kk + k] = ti[k];
        }
    }
}

// -------------------------------------------------------------------------
// Kernel 2: query[32,1024] = inputs @ Wq^T + bq   (WMMA f32 16x16x4)
//   B[k][n] = Wq[n][k]  -> contiguous float2 per lane along k
// -------------------------------------------------------------------------
__global__ void k_query(const float* __restrict__ inp,
                        const float* __restrict__ Wq,
                        const float* __restrict__ bq,
                        float* __restrict__ query) {
    const int lane = threadIdx.x & 31;
    const int wave = threadIdx.x >> 5;
    const int tile = blockIdx.x * 4 + wave;        // 128 tiles: 2 (M) x 64 (N)
    const int m0 = (tile & 1) * 16;
    const int n0 = (tile >> 1) * 16;
    const int hi = lane >> 4;                      // 0/1 -> K offset 0/2
    const int li = lane & 15;

    v8f c = {};
    const float* arow = inp + (size_t)(m0 + li) * Ee;
    const float* brow = Wq + (size_t)(n0 + li) * Ee;
    for (int k0 = 0; k0 < Ee; k0 += 4) {
        const v2f a = *(const v2f*)(arow + k0 + 2 * hi);
        const v2f bf = *(const v2f*)(brow + k0 + 2 * hi);
        c = __builtin_amdgcn_wmma_f32_16x16x4_f32(false, a, false, bf, (short)0, c, false, false);
    }
    const int col = n0 + li;
    const float bias = bq[col];
#pragma unroll
    for (int r = 0; r < 8; ++r) {
        const int row = m0 + r + 8 * hi;
        query[(size_t)row * Hh + col] = c[r] + bias;
    }
}

// -------------------------------------------------------------------------
// Kernel 3: q2[32,1024] = (query @ Wk) * (1/sqrt(H))   (WMMA f32 16x16x4)
//   B[k][n] = Wk[k][n]  -> lane-coalesced b32 loads
// -------------------------------------------------------------------------
__global__ void k_q2(const float* __restrict__ query,
                     const float* __restrict__ Wk,
                     float* __restrict__ q2) {
    const int lane = threadIdx.x & 31;
    const int wave = threadIdx.x >> 5;
    const int tile = blockIdx.x * 4 + wave;
    const int m0 = (tile & 1) * 16;
    const int n0 = (tile >> 1) * 16;
    const int hi = lane >> 4;
    const int li = lane & 15;

    v8f c = {};
    const float* arow = query + (size_t)(m0 + li) * Hh;
    const int col = n0 + li;
    for (int k0 = 0; k0 < Hh; k0 += 4) {
        const v2f a = *(const v2f*)(arow + k0 + 2 * hi);
        const int kk = k0 + 2 * hi;
        v2f bf;
        bf.x = Wk[(size_t)kk * Hh + col];
        bf.y = Wk[(size_t)(kk + 1) * Hh + col];
        c = __builtin_amdgcn_wmma_f32_16x16x4_f32(false, a, false, bf, (short)0, c, false, false);
    }
    const float inv_sqrt = 0.03125f;  // 1/sqrt(1024)
#pragma unroll
    for (int r = 0; r < 8; ++r) {
        const int row = m0 + r + 8 * hi;
        q2[(size_t)row * Hh + col] = c[r] * inv_sqrt;
    }
}

// -------------------------------------------------------------------------
// Kernel 4: per (b,k) zone: logits -> softmax -> weighted zone sum
//   pout[b,k,h] = weights[b,k] * softmax_l(zone.q2)[l] * zone[l,h]
// -------------------------------------------------------------------------
__global__ void k_attn(const float* __restrict__ cv,
                       const float* __restrict__ q2,
                       const float* __restrict__ wts,
                       const int* __restrict__ tidx,
                       float* __restrict__ pout) {
    __shared__ float s_q[Hh];
    __shared__ float s_lg[Ll];
    __shared__ float s_cf[Ll];

    const int bkid = blockIdx.x;           // 0..255
    const int b = bkid >> 3;
    const int n_sel = tidx[bkid];
    const float* zone = cv + (((size_t)b * Nn + n_sel) * Ll) * Hh;

    for (int h = threadIdx.x; h < Hh; h += 256) s_q[h] = q2[b * Hh + h];
    __syncthreads();

    const int lane = threadIdx.x & 31;
    const int wave = threadIdx.x >> 5;
    for (int l = wave; l < Ll; l += 8) {
        const float* row = zone + (size_t)l * Hh;
        float p = 0.f;
#pragma unroll
        for (int j = 0; j < 8; ++j) {
            const int e = lane * 4 + j * 128;
            const float4 v = *(const float4*)(row + e);
            p += v.x * s_q[e] + v.y * s_q[e + 1] + v.z * s_q[e + 2] + v.w * s_q[e + 3];
        }
        p = wave_reduce_sum(p);
        if (lane == 0) s_lg[l] = p;
    }
    __syncthreads();

    if (threadIdx.x == 0) {
        float m = -INFINITY;
        for (int l = 0; l < Ll; ++l) m = fmaxf(m, s_lg[l]);
        float s = 0.f;
        for (int l = 0; l < Ll; ++l) { const float e = expf(s_lg[l] - m); s_cf[l] = e; s += e; }
        const float w = wts[bkid] / s;
        for (int l = 0; l < Ll; ++l) s_cf[l] *= w;
    }
    __syncthreads();

    const int t = threadIdx.x;
    float a0 = 0.f, a1 = 0.f, a2 = 0.f, a3 = 0.f;
    for (int l = 0; l < Ll; ++l) {
        const float cf = s_cf[l];
        const float* row = zone + (size_t)l * Hh;
        a0 += cf * row[t];
        a1 += cf * row[t + 256];
        a2 += cf * row[t + 512];
        a3 += cf * row[t + 768];
    }
    float* po = pout + (size_t)bkid * Hh;
    po[t] = a0; po[t + 256] = a1; po[t + 512] = a2; po[t + 768] = a3;
}

// -------------------------------------------------------------------------
// Kernel 4b: attn_out[b,h] = sum_k pout[b,k,h]
// -------------------------------------------------------------------------
__global__ void k_reduce_k(const float* __restrict__ pout, float* __restrict__ aout) {
    const int i = blockIdx.x * 256 + threadIdx.x;   // 0..32767
    const int b = i >> 10;
    const int h = i & 1023;
    const float* p = pout + (size_t)b * Kk * Hh + h;
    float s = 0.f;
#pragma unroll
    for (int k = 0; k < Kk; ++k) s += p[k * Hh];
    aout[i] = s;
}

// -------------------------------------------------------------------------
// Kernel 5: out[32,1024] = [attn_out, inputs] @ Wc^T + bc  (WMMA f32 16x16x4)
//   split K=2048 into two 1024 halves; B[k][n] = Wc[n][k] / Wc[n][H+k]
// -------------------------------------------------------------------------
__global__ void k_final(const float* __restrict__ aout,
                        const float* __restrict__ inp,
                        const float* __restrict__ Wc,
                        const float* __restrict__ bc,
                        float* __restrict__ out) {
    const int lane = threadIdx.x & 31;
    const int wave = threadIdx.x >> 5;
    const int tile = blockIdx.x * 4 + wave;
    const int m0 = (tile & 1) * 16;
    const int n0 = (tile >> 1) * 16;
    const int hi = lane >> 4;
    const int li = lane & 15;

    v8f c = {};
    const float* arow0 = aout + (size_t)(m0 + li) * Hh;
    const float* arow1 = inp + (size_t)(m0 + li) * Ee;
    const float* brow = Wc + (size_t)(n0 + li) * (2 * Hh);
    for (int k0 = 0; k0 < Hh; k0 += 4) {
        const v2f a = *(const v2f*)(arow0 + k0 + 2 * hi);
        const v2f bf = *(const v2f*)(brow + k0 + 2 * hi);
        c = __builtin_amdgcn_wmma_f32_16x16x4_f32(false, a, false, bf, (short)0, c, false, false);
    }
    for (int k0 = 0; k0 < Hh; k0 += 4) {
        const v2f a = *(const v2f*)(arow1 + k0 + 2 * hi);
        const v2f bf = *(const v2f*)(brow + Hh + k0 + 2 * hi);
        c = __builtin_amdgcn_wmma_f32_16x16x4_f32(false, a, false, bf, (short)0, c, false, false);
    }
    const int col = n0 + li;
    const float bias = bc[col];
#pragma unroll
    for (int r = 0; r < 8; ++r) {
        const int row = m0 + r + 8 * hi;
        out[(size_t)row * Hh + col] = c[r] + bias;
    }
}

// -------------------------------------------------------------------------
extern "C" void kernel_launch(void* const* d_in, const int* in_sizes, int n_in,
                              void* d_out, int out_size, void* d_ws, size_t ws_size,
                              hipStream_t stream) {
    const float* inp = (const float*)d_in[0];   // [B,E]
    const float* Wq  = (const float*)d_in[1];   // [H,E]
    const float* bq  = (const float*)d_in[2];   // [H]
    const float* Wk  = (const float*)d_in[3];   // [H,H]
    // d_in[4] = bk : provably cancels inside softmax over L (constant shift)
    const float* Wc  = (const float*)d_in[5];   // [H,2H]
    const float* bc  = (const float*)d_in[6];   // [H]
    const float* ck  = (const float*)d_in[7];   // [B,N,E]
    const float* cv  = (const float*)d_in[8];   // [B,N,L,H]
    float* out = (float*)d_out;                 // [B,H]

    // workspace carve-up (bytes)
    char* ws = (char*)d_ws;
    float* wts   = (float*)(ws + 0);            //  256 f
    int*   tidx  = (int*)  (ws + 1024);         //  256 i
    float* query = (float*)(ws + 2048);         //  32768 f
    float* q2    = (float*)(ws + 2048 + 131072);            // 32768 f
    float* pout  = (float*)(ws + 2048 + 2 * 131072);        // 262144 f
    float* aout  = (float*)(ws + 2048 + 2 * 131072 + 1048576); // 32768 f

    k_scores_topk<<<Bb, 256, 0, stream>>>(inp, ck, wts, tidx);
    k_query<<<32, 128, 0, stream>>>(inp, Wq, bq, query);
    k_q2<<<32, 128, 0, stream>>>(query, Wk, q2);
    k_attn<<<Bb * Kk, 256, 0, stream>>>(cv, q2, wts, tidx, pout);
    k_reduce_k<<<(Bb * Hh) / 256, 256, 0, stream>>>(pout, aout);
    k_final<<<32, 128, 0, stream>>>(aout, inp, Wc, bc, out);
}